// SageNet_34849364640442
// MI455X (gfx1250) — compile-verified
//
#include <hip/hip_runtime.h>
#include <hip/hip_bf16.h>
#include <stdint.h>

// ---------------------------------------------------------------------------
// GraphSAGE 2-layer inference for MI455X (gfx1250).
//
// Project-then-aggregate: mean-aggregation commutes with the linear layers,
// so we project to 48-wide first and aggregate the 19.2MB projected features
// entirely inside the 192MB L2. Both projections (lin_l, lin_r) are fused in
// one WMMA kernel (shared A fragments, two accumulators), with the weight
// slices staged into LDS via gfx1250 async-to-LDS loads (ASYNCcnt) and read
// back as conflict-free ds_load_b64 fragments. GEMMs run on the fp32 matrix
// pipe (V_WMMA_F32_16X16X4_F32) to match the fp32 reference exactly.
// HBM traffic ~ x(102MB) + edges(2x25.6MB) + out(1.6MB) -> ~6-8us floor
// at 23.3 TB/s; the ~5 GFLOP of matrix work is negligible on the WMMA pipe.
// ---------------------------------------------------------------------------

typedef __attribute__((ext_vector_type(2))) float v2f;
typedef __attribute__((ext_vector_type(4))) float v4f;
typedef __attribute__((ext_vector_type(8))) float v8f;

// Types matching the async-to-LDS builtin's parameters:
//   param0: AS1 (global / __device__) pointer to 16B int vector
//   param1: AS3 (LDS / __shared__)   pointer to 16B int vector
typedef int v4i_g __attribute__((vector_size(16)));
typedef __attribute__((address_space(1))) v4i_g* as1_v4i_ptr;
typedef __attribute__((address_space(3))) v4i_g* as3_v4i_ptr;

#define HIDF 48

// ---- gfx1250 async global->LDS copy helpers (compile-safe fallbacks) ------
__device__ __forceinline__ void async_copy_f4(const float* __restrict__ g, float* l) {
#if defined(__has_builtin) && __has_builtin(__builtin_amdgcn_global_load_async_to_lds_b128)
  // (AS1 v4i*, AS3 v4i*, imm offset, imm cpol) -> global_load_async_to_lds_b128
  __builtin_amdgcn_global_load_async_to_lds_b128(
      (as1_v4i_ptr)(uintptr_t)g,
      (as3_v4i_ptr)(uint32_t)(uintptr_t)l, 0, 0);
#else
  *(v4f*)l = *(const v4f*)g;
#endif
}

__device__ __forceinline__ void wait_async_all() {
#if defined(__has_builtin) && __has_builtin(__builtin_amdgcn_s_wait_asynccnt)
  __builtin_amdgcn_s_wait_asynccnt(0);
#else
  asm volatile("s_wait_asynccnt 0x0" ::: "memory");
#endif
}

// ---------------------------------------------------------------------------
// Fused dual projection: Yl = X @ Wl^T, Yr = X @ Wr^T  (X: n x K, W*: F x K).
// grid = (ceil(rowTiles/8), F/16); 8 waves/block, one 16x16 tile per wave,
// all waves in a block share the col-tile so the two 16xK weight slices are
// staged into LDS once per block (async DMA), then consumed as ds_load_b64.
// Fragment layouts per CDNA5 ISA 7.12.2. Requires K%4==0, n%16==0, F%16==0.
// ---------------------------------------------------------------------------
__global__ void gemm_dual_xwT_wmma(const float* __restrict__ X,
                                   const float* __restrict__ Wl,
                                   const float* __restrict__ Wr,
                                   float* __restrict__ Yl,
                                   float* __restrict__ Yr,
                                   int K, int F, int rowTilesTotal) {
  extern __shared__ float lds[];
  const int stride = K + 4;                 // pad: 16 distinct banks per half-wave
  float* ldsL = lds;
  float* ldsR = lds + 16 * stride;

  const int col0 = (int)blockIdx.y << 4;
  const int tid  = threadIdx.x;

  // ---- stage Wl/Wr 16xK slices into LDS (contiguous 16*K floats each) ----
  const int f4PerMat = 16 * (K >> 2);
  const float* gWl = Wl + (size_t)col0 * K;
  const float* gWr = Wr + (size_t)col0 * K;
  for (int j = tid; j < f4PerMat; j += 256) {
    int r  = j / (K >> 2);
    int kk = (j - r * (K >> 2)) << 2;       // float offset within row
    async_copy_f4(gWl + r * K + kk, ldsL + r * stride + kk);
    async_copy_f4(gWr + r * K + kk, ldsR + r * stride + kk);
  }
  wait_async_all();
  __syncthreads();

  const int wave    = tid >> 5;
  const int lane    = tid & 31;
  const int rowTile = (int)blockIdx.x * 8 + wave;
  if (rowTile >= rowTilesTotal) return;     // wave-uniform: EXEC stays all-ones

  const int row0 = rowTile << 4;
  const int half = lane >> 4;               // 0: lanes 0-15, 1: lanes 16-31
  const int l15  = lane & 15;

  const float* xrow = X + (size_t)(row0 + l15) * K + 2 * half;   // A: row M=l15
  const float* blp  = ldsL + l15 * stride + 2 * half;            // B: col N=l15
  const float* brp  = ldsR + l15 * stride + 2 * half;

  v8f cl = {};
  v8f cr = {};
  for (int k = 0; k < K; k += 4) {
    v2f a  = *(const v2f*)(xrow + k);
    v2f bl = *(const v2f*)(blp + k);
    v2f br = *(const v2f*)(brp + k);
    cl = __builtin_amdgcn_wmma_f32_16x16x4_f32(false, a, false, bl, (short)0, cl,
                                               false, false);
    cr = __builtin_amdgcn_wmma_f32_16x16x4_f32(false, a, false, br, (short)0, cr,
                                               false, false);
  }

  // D layout: VGPR r -> (M = r + 8*half, N = l15)
  const size_t obase = (size_t)(row0 + half * 8) * F + col0 + l15;
  float* yl = Yl + obase;
  float* yr = Yr + obase;
#pragma unroll
  for (int r = 0; r < 8; ++r) {
    yl[(size_t)r * F] = cl[r];
    yr[(size_t)r * F] = cr[r];
  }
}

__global__ void zero_f32(float* __restrict__ p, long long count) {
  long long i = (long long)blockIdx.x * blockDim.x + threadIdx.x;
  if (i < count) p[i] = 0.0f;
}

// Scatter-sum of projected features along edges. 64 threads per edge:
// lanes 0-47 add one channel each (consecutive lanes -> consecutive addrs,
// global_atomic_add_f32 into the L2-resident 19.2MB agg buffer), lane 48
// bumps the degree, lane 49 prefetches an upcoming source row.
__global__ void edge_aggregate(const float* __restrict__ xl,
                               const int* __restrict__ src,
                               const int* __restrict__ dst,
                               float* __restrict__ agg,
                               float* __restrict__ deg, int E) {
  long long t = (long long)blockIdx.x * blockDim.x + threadIdx.x;
  int e = (int)(t >> 6);
  int c = (int)(t & 63);
  if (e >= E) return;
  int s = src[e];
  int d = dst[e];
  if (c < HIDF) {
    atomicAdd(&agg[(size_t)d * HIDF + c], xl[(size_t)s * HIDF + c]);
  } else if (c == 48) {
    atomicAdd(&deg[d], 1.0f);
  } else if (c == 49 && e + 4096 < E) {
    __builtin_prefetch(&xl[(size_t)src[e + 4096] * HIDF], 0, 1);
  }
}

// h = relu( bn( agg/max(deg,1) + bl + xr ) )
__global__ void combine_bn_relu(const float* __restrict__ agg,
                                const float* __restrict__ deg,
                                const float* __restrict__ xr,
                                const float* __restrict__ bl,
                                const float* __restrict__ g,
                                const float* __restrict__ be,
                                const float* __restrict__ rm,
                                const float* __restrict__ rv,
                                float* __restrict__ h, int n) {
  int t = blockIdx.x * blockDim.x + threadIdx.x;
  if (t >= n * HIDF) return;
  int i  = t / HIDF;
  int ch = t - i * HIDF;
  float dg = deg[i];
  dg = dg > 1.0f ? dg : 1.0f;
  float v = agg[t] / dg + bl[ch] + xr[t];
  v = (v - rm[ch]) * (g[ch] * rsqrtf(rv[ch] + 1e-5f)) + be[ch];
  h[t] = v > 0.0f ? v : 0.0f;
}

// out[N x 4] = h[N x 48] @ wo[4 x 48]^T + bo   (too skinny for WMMA; mem-bound)
__global__ void out_linear(const float* __restrict__ H,
                           const float* __restrict__ WO,
                           const float* __restrict__ BO,
                           float* __restrict__ out, int n) {
  int i = blockIdx.x * blockDim.x + threadIdx.x;
  if (i >= n) return;
  const float* h = H + (size_t)i * HIDF;
  float acc0 = BO[0], acc1 = BO[1], acc2 = BO[2], acc3 = BO[3];
#pragma unroll 8
  for (int k = 0; k < HIDF; ++k) {
    float hv = h[k];
    acc0 += hv * WO[0 * HIDF + k];
    acc1 += hv * WO[1 * HIDF + k];
    acc2 += hv * WO[2 * HIDF + k];
    acc3 += hv * WO[3 * HIDF + k];
  }
  float* o = out + (size_t)i * 4;
  o[0] = acc0; o[1] = acc1; o[2] = acc2; o[3] = acc3;
}

extern "C" void kernel_launch(void* const* d_in, const int* in_sizes, int n_in,
                              void* d_out, int out_size, void* d_ws, size_t ws_size,
                              hipStream_t stream) {
  const float* x   = (const float*)d_in[0];
  const int*   ei  = (const int*)d_in[1];
  const float* w1l = (const float*)d_in[2];
  const float* b1  = (const float*)d_in[3];
  const float* w1r = (const float*)d_in[4];
  const float* g1  = (const float*)d_in[5];
  const float* be1 = (const float*)d_in[6];
  const float* rm1 = (const float*)d_in[7];
  const float* rv1 = (const float*)d_in[8];
  const float* w2l = (const float*)d_in[9];
  const float* b2  = (const float*)d_in[10];
  const float* w2r = (const float*)d_in[11];
  const float* g2  = (const float*)d_in[12];
  const float* be2 = (const float*)d_in[13];
  const float* rm2 = (const float*)d_in[14];
  const float* rv2 = (const float*)d_in[15];
  const float* wo  = (const float*)d_in[16];
  const float* bo  = (const float*)d_in[17];

  const int IN_CH = 256;
  const int N = in_sizes[0] / IN_CH;     // 100000 (divisible by 16)
  const int E = in_sizes[1] / 2;         // 1600000
  const int* srcIdx = ei;
  const int* dstIdx = ei + E;

  // Workspace layout (floats): xl | xr | agg | deg | h1   (~77.2 MB)
  float* ws  = (float*)d_ws;
  size_t n48 = (size_t)N * HIDF;
  float* xl  = ws;
  float* xr  = ws + n48;
  float* agg = ws + 2 * n48;
  float* deg = ws + 3 * n48;
  float* h1  = ws + 3 * n48 + N;
  float* outp = (float*)d_out;

  const int rowTiles = N / 16;                                  // 6250
  const dim3 gGrd((rowTiles + 7) / 8, HIDF / 16, 1);            // 782 x 3
  const size_t shm1 = (size_t)2 * 16 * (IN_CH + 4) * sizeof(float);  // 33.3 KB
  const size_t shm2 = (size_t)2 * 16 * (HIDF + 4) * sizeof(float);   //  6.7 KB

  const long long zCount = (long long)n48 + N;                  // agg+deg contiguous
  const int zGrd = (int)((zCount + 255) / 256);

  const long long eThreads = (long long)E * 64;
  const int eGrd = (int)((eThreads + 255) / 256);

  const int cGrd = (int)(((long long)N * HIDF + 255) / 256);
  const int oGrd = (N + 255) / 256;

  // ---- Layer 1 ----
  gemm_dual_xwT_wmma<<<gGrd, 256, shm1, stream>>>(x, w1l, w1r, xl, xr,
                                                  IN_CH, HIDF, rowTiles);
  zero_f32<<<zGrd, 256, 0, stream>>>(agg, zCount);
  edge_aggregate<<<eGrd, 256, 0, stream>>>(xl, srcIdx, dstIdx, agg, deg, E);
  combine_bn_relu<<<cGrd, 256, 0, stream>>>(agg, deg, xr, b1, g1, be1, rm1, rv1,
                                            h1, N);

  // ---- Layer 2 ----
  gemm_dual_xwT_wmma<<<gGrd, 256, shm2, stream>>>(h1, w2l, w2r, xl, xr,
                                                  HIDF, HIDF, rowTiles);
  zero_f32<<<zGrd, 256, 0, stream>>>(agg, zCount);
  edge_aggregate<<<eGrd, 256, 0, stream>>>(xl, srcIdx, dstIdx, agg, deg, E);
  combine_bn_relu<<<cGrd, 256, 0, stream>>>(agg, deg, xr, b2, g2, be2, rm2, rv2,
                                            xl, N);   // h2 -> xl buffer (free now)

  // ---- Output head ----
  out_linear<<<oGrd, 256, 0, stream>>>(xl, wo, bo, outp, N);
}